// NSHE_87943750353447
// MI455X (gfx1250) — compile-verified
//
#include <hip/hip_runtime.h>
#include <hip/hip_bf16.h>

typedef __attribute__((ext_vector_type(2))) float v2f;
typedef __attribute__((ext_vector_type(8))) float v8f;

#define N_A    50000
#define N_P    30000
#define N_S    20000
#define NTOT   (N_A + N_P + N_S)      // 100000
#define IN_DIM 128
#define PROJ   64
#define EMD    64
#define CTXW   32
#define HID    16
#define SMP    100000
#define NEDGE  3200000
#define CLA    (EMD + 2 * CTXW)       // 128

// ---------------------------------------------------------------------------
// WMMA f32 16x16x4 tile helper.  ISA 7.12.2 layouts (wave32):
//   A 16x4:  lanes 0-15 -> M=lane, VGPR pair holds K=kb+0,kb+1
//            lanes 16-31 -> M=lane-16, K=kb+2,kb+3
//   B 4x16:  lanes 0-15 -> N=lane rows K=kb+0,kb+1 ; lanes 16-31 K=kb+2,kb+3
//   C 16x16: VGPR r -> M=r (lanes 0-15) / M=r+8 (lanes 16-31), N=lane&15
// A comes from LDS (ds_load_b64 per fragment), B through the cache hierarchy
// (weights are tiny and L2/L0 resident).
// ---------------------------------------------------------------------------
__device__ __forceinline__ v8f wmma_tile_f32(const float* ldsA, int lda,
                                             const float* __restrict__ B, int ldb,
                                             int ncol_base, int K, v8f acc)
{
    const int lane = threadIdx.x & 31;
    const int half = lane >> 4;
    const int l15  = lane & 15;
    const int ncol = ncol_base + l15;
    for (int kb = 0; kb < K; kb += 4) {
        const int k0 = kb + half * 2;
        v2f a = *(const v2f*)(ldsA + l15 * lda + k0);
        v2f b;
        b.x = B[(k0 + 0) * ldb + ncol];
        b.y = B[(k0 + 1) * ldb + ncol];
        acc = __builtin_amdgcn_wmma_f32_16x16x4_f32(false, a, false, b,
                                                    (short)0, acc, false, false);
    }
    return acc;
}

__device__ __forceinline__ void store_tile(float* __restrict__ out, int ldo,
                                           int row_base, int ncol_base, v8f c)
{
    const int lane = threadIdx.x & 31;
    const int half = lane >> 4;
    const int l15  = lane & 15;
#pragma unroll
    for (int r = 0; r < 8; ++r)
        out[(size_t)(row_base + r + half * 8) * ldo + ncol_base + l15] = c[r];
}

// --------------------------- zero workspace --------------------------------
__global__ __launch_bounds__(256) void k_zero(float4* p, size_t n4)
{
    size_t i = (size_t)blockIdx.x * blockDim.x + threadIdx.x;
    if (i < n4) p[i] = float4{0.f, 0.f, 0.f, 0.f};
}

// ------------------- per-type projection: hp = H @ Wproj -------------------
// block = 128 threads = 4 waves; block does 16 rows x 64 cols (wave w: cols 16w..)
__global__ __launch_bounds__(128) void k_proj(const float* __restrict__ H,
                                              const float* __restrict__ W,
                                              float* __restrict__ hp, int out_row_off)
{
    __shared__ float ldsA[16 * IN_DIM];
    const int rb = blockIdx.x * 16;
    for (int i = threadIdx.x; i < 16 * IN_DIM; i += 128)
        ldsA[i] = H[(size_t)(rb + (i >> 7)) * IN_DIM + (i & 127)];
    __syncthreads();
    const int wave = threadIdx.x >> 5;
    v8f c = {};
    c = wmma_tile_f32(ldsA, IN_DIM, W, PROJ, wave * 16, IN_DIM, c);
    store_tile(hp + (size_t)out_row_off * PROJ, PROJ, rb, wave * 16, c);
}

// ------------------- edge scatter: agg[dst] += hp[src] ---------------------
// 16 lanes per edge, float4 each -> one coalesced 256B row read per edge;
// native global_atomic_add_f32 (relaxed, device scope) at L2 (hp/agg L2-resident).
__global__ __launch_bounds__(256) void k_edges(const int* __restrict__ src,
                                               const int* __restrict__ dst,
                                               const float* __restrict__ hp,
                                               float* __restrict__ agg)
{
    const long long tid = (long long)blockIdx.x * 256 + threadIdx.x;
    const long long e = tid >> 4;
    if (e >= NEDGE) return;
    const int c = (int)(tid & 15) << 2;
    const int s = src[e], d = dst[e];
    const float4 v = *(const float4*)(hp + (size_t)s * PROJ + c);
    float* ap = agg + (size_t)d * PROJ + c;
    __hip_atomic_fetch_add(ap + 0, v.x, __ATOMIC_RELAXED, __HIP_MEMORY_SCOPE_AGENT);
    __hip_atomic_fetch_add(ap + 1, v.y, __ATOMIC_RELAXED, __HIP_MEMORY_SCOPE_AGENT);
    __hip_atomic_fetch_add(ap + 2, v.z, __ATOMIC_RELAXED, __HIP_MEMORY_SCOPE_AGENT);
    __hip_atomic_fetch_add(ap + 3, v.w, __ATOMIC_RELAXED, __HIP_MEMORY_SCOPE_AGENT);
}

// ------------- h = rownorm(agg @ W2 + b2), fused, writes d_out -------------
__global__ __launch_bounds__(128) void k_h(const float* __restrict__ agg,
                                           const float* __restrict__ W2,
                                           const float* __restrict__ b2,
                                           float* __restrict__ hout)
{
    __shared__ float ldsA[16 * EMD];
    __shared__ float ldsC[16 * EMD];
    __shared__ float rnorm[16];
    const int rb = blockIdx.x * 16;
    for (int i = threadIdx.x; i < 16 * EMD; i += 128)
        ldsA[i] = agg[(size_t)(rb + (i >> 6)) * EMD + (i & 63)];
    __syncthreads();
    const int wave = threadIdx.x >> 5;
    v8f c = {};
    c = wmma_tile_f32(ldsA, EMD, W2, EMD, wave * 16, EMD, c);
    {
        const int lane = threadIdx.x & 31;
        const int half = lane >> 4, l15 = lane & 15;
        const int col = wave * 16 + l15;
        const float bias = b2[col];
#pragma unroll
        for (int r = 0; r < 8; ++r)
            ldsC[(r + half * 8) * EMD + col] = c[r] + bias;
    }
    __syncthreads();
    if (threadIdx.x < 16) {
        float s = 0.f;
        for (int j = 0; j < EMD; ++j) { float v = ldsC[threadIdx.x * EMD + j]; s += v * v; }
        rnorm[threadIdx.x] = 1.f / fmaxf(sqrtf(s), 1e-12f);
    }
    __syncthreads();
    for (int i = threadIdx.x; i < 16 * EMD; i += 128)
        hout[(size_t)(rb + (i >> 6)) * EMD + (i & 63)] = ldsC[i] * rnorm[i >> 6];
}

// ---------------- ctx = h @ Wc_type (type boundaries 16-aligned) -----------
__global__ __launch_bounds__(64) void k_ctx(const float* __restrict__ h,
                                            const float* __restrict__ Wc_a,
                                            const float* __restrict__ Wc_p,
                                            const float* __restrict__ Wc_s,
                                            float* __restrict__ ctx)
{
    __shared__ float ldsA[16 * EMD];
    const int rb = blockIdx.x * 16;
    const float* W = (rb < N_A) ? Wc_a : (rb < N_A + N_P) ? Wc_p : Wc_s;
    for (int i = threadIdx.x; i < 16 * EMD; i += 64)
        ldsA[i] = h[(size_t)(rb + (i >> 6)) * EMD + (i & 63)];
    __syncthreads();
    const int wave = threadIdx.x >> 5;
    v8f c = {};
    c = wmma_tile_f32(ldsA, EMD, W, CTXW, wave * 16, EMD, c);
    store_tile(ctx, CTXW, rb, wave * 16, c);
}

// -------- head: x = sigmoid(relu(gather(feat) @ Wh) @ Wo), per type --------
// block = 128 threads = 4 waves; 64 gathered rows (16 per wave), K = 128.
__global__ __launch_bounds__(128) void k_head(const float* __restrict__ h,
                                              const float* __restrict__ ctx,
                                              const float* __restrict__ Wh,
                                              const float* __restrict__ Wo,
                                              const int* __restrict__ idx_self,
                                              const int* __restrict__ idx_c1,
                                              const int* __restrict__ idx_c2,
                                              int off_self, int off_c1, int off_c2,
                                              float* __restrict__ xout)
{
    __shared__ float ldsA[64 * CLA];                 // 32 KB
    const int rb = blockIdx.x * 64;
    {
        const int r = threadIdx.x >> 1;
        const int i = rb + r;
        if (i < SMP) {
            float* drow = ldsA + r * CLA;
            if ((threadIdx.x & 1) == 0) {
                const float* s0 = h + (size_t)(off_self + idx_self[i]) * EMD;
#pragma unroll
                for (int j = 0; j < EMD / 4; ++j)
                    *(float4*)(drow + j * 4) = *(const float4*)(s0 + j * 4);
            } else {
                const float* s1 = ctx + (size_t)(off_c1 + idx_c1[i]) * CTXW;
                const float* s2 = ctx + (size_t)(off_c2 + idx_c2[i]) * CTXW;
#pragma unroll
                for (int j = 0; j < CTXW / 4; ++j) {
                    *(float4*)(drow + EMD + j * 4)        = *(const float4*)(s1 + j * 4);
                    *(float4*)(drow + EMD + CTXW + j * 4) = *(const float4*)(s2 + j * 4);
                }
            }
        }
    }
    __syncthreads();
    const int wave = threadIdx.x >> 5;
    const int trow = rb + wave * 16;                 // wave-uniform guard: EXEC all-1s
    if (trow < SMP) {
        v8f c = {};
        c = wmma_tile_f32(ldsA + wave * 16 * CLA, CLA, Wh, HID, 0, CLA, c);
        const int lane = threadIdx.x & 31;
        const int l15 = lane & 15, half = lane >> 4;
        const float wo = Wo[l15];
#pragma unroll
        for (int r = 0; r < 8; ++r) {
            float v = fmaxf(c[r], 0.f) * wo;         // relu + Wo weight, col = l15
            v += __shfl_xor(v, 1, 32);
            v += __shfl_xor(v, 2, 32);
            v += __shfl_xor(v, 4, 32);
            v += __shfl_xor(v, 8, 32);               // sum over 16 cols (half-wave)
            if (l15 == 0)
                xout[trow + r + half * 8] = 1.f / (1.f + expf(-v));
        }
    }
}

// ---------------------------------------------------------------------------
extern "C" void kernel_launch(void* const* d_in, const int* in_sizes, int n_in,
                              void* d_out, int out_size, void* d_ws, size_t ws_size,
                              hipStream_t stream)
{
    (void)in_sizes; (void)n_in; (void)out_size; (void)ws_size;
    const float* h_a  = (const float*)d_in[0];
    const float* h_p  = (const float*)d_in[1];
    const float* h_s  = (const float*)d_in[2];
    const float* Wpa  = (const float*)d_in[3];
    const float* Wpp  = (const float*)d_in[4];
    const float* Wps  = (const float*)d_in[5];
    const float* W2   = (const float*)d_in[6];
    const float* b2   = (const float*)d_in[7];
    const float* Wc_a = (const float*)d_in[8];
    const float* Wc_p = (const float*)d_in[9];
    const float* Wc_s = (const float*)d_in[10];
    const float* Wh_a = (const float*)d_in[11];
    const float* Wh_p = (const float*)d_in[12];
    const float* Wh_s = (const float*)d_in[13];
    const float* Wo_a = (const float*)d_in[14];
    const float* Wo_p = (const float*)d_in[15];
    const float* Wo_s = (const float*)d_in[16];
    const int*   src  = (const int*)d_in[17];
    const int*   dst  = (const int*)d_in[18];
    const int* ns_a_a = (const int*)d_in[19];
    const int* ns_a_p = (const int*)d_in[20];
    const int* ns_a_s = (const int*)d_in[21];
    const int* ns_p_a = (const int*)d_in[22];
    const int* ns_p_p = (const int*)d_in[23];
    const int* ns_p_s = (const int*)d_in[24];
    const int* ns_s_a = (const int*)d_in[25];
    const int* ns_s_p = (const int*)d_in[26];
    const int* ns_s_s = (const int*)d_in[27];

    float* out_h = (float*)d_out;                       // [NTOT*64]
    float* out_x = out_h + (size_t)NTOT * EMD;          // [3*SMP]

    // workspace: hp (25.6MB) | agg (25.6MB); ctx reuses hp region after edges
    float* hp  = (float*)d_ws;
    float* agg = hp + (size_t)NTOT * PROJ;
    float* ctx = hp;

    // 1) zero accumulator (must re-zero every call; graph replays)
    const size_t n4 = (size_t)NTOT * PROJ / 4;
    k_zero<<<(unsigned)((n4 + 255) / 256), 256, 0, stream>>>((float4*)agg, n4);

    // 2) projections (row counts all multiples of 16)
    k_proj<<<N_A / 16, 128, 0, stream>>>(h_a, Wpa, hp, 0);
    k_proj<<<N_P / 16, 128, 0, stream>>>(h_p, Wpp, hp, N_A);
    k_proj<<<N_S / 16, 128, 0, stream>>>(h_s, Wps, hp, N_A + N_P);

    // 3) edge scatter-add (L2-resident hp/agg, native f32 atomics)
    k_edges<<<(NEDGE * 16) / 256, 256, 0, stream>>>(src, dst, hp, agg);

    // 4) h = rownorm(agg @ W2 + b2) -> d_out
    k_h<<<NTOT / 16, 128, 0, stream>>>(agg, W2, b2, out_h);

    // 5) ctx = h @ Wc_type (overwrites hp region)
    k_ctx<<<NTOT / 16, 64, 0, stream>>>(out_h, Wc_a, Wc_p, Wc_s, ctx);

    // 6) heads; feature order per reference: [emb_t, ctx_other1, ctx_other2] (a,p,s order)
    const int hb = (SMP + 63) / 64;
    k_head<<<hb, 128, 0, stream>>>(out_h, ctx, Wh_a, Wo_a, ns_a_a, ns_a_p, ns_a_s,
                                   0, N_A, N_A + N_P, out_x);
    k_head<<<hb, 128, 0, stream>>>(out_h, ctx, Wh_p, Wo_p, ns_p_p, ns_p_a, ns_p_s,
                                   N_A, 0, N_A + N_P, out_x + SMP);
    k_head<<<hb, 128, 0, stream>>>(out_h, ctx, Wh_s, Wo_s, ns_s_s, ns_s_a, ns_s_p,
                                   N_A + N_P, 0, N_A, out_x + 2 * SMP);
}